// EquivarianceLoss_42039139893371
// MI455X (gfx1250) — compile-verified
//
#include <hip/hip_runtime.h>
#include <hip/hip_bf16.h>
#include <stddef.h>

// EquivarianceLoss for MI455X / gfx1250 (wave32, WMMA, async-LDS).
// [1] 256 blocks: fused log/entropy + batched 8x(HW)x8 cross-GEMM partials via
//     V_WMMA_F32_16X16X4_F32 (2 batches per 16x16 tile, HW split 8-ways),
//     aug tiles staged with GLOBAL_LOAD_ASYNC_TO_LDS_B32 when available.
// [2] deterministic slice combine -> C.
// [3] exact assignment over 8! perms (on-device Lehmer decode).
// [4] permute rows + KL partials, [5] scalar reduce.
// All reductions are tree/serial in LDS/registers -> bit-deterministic.

#define EPSF 1e-15f

typedef __attribute__((ext_vector_type(2))) float v2f;
typedef __attribute__((ext_vector_type(8))) float v8f;

// Builtin async-load pointer types: global src is AS1 ("__device__ int *"
// per clang's diagnostic), LDS dst is AS3.
typedef __attribute__((address_space(1))) int gas_int;
typedef __attribute__((address_space(3))) int las_int;

constexpr int Bn     = 64;
constexpr int Kn     = 8;
constexpr int HWn    = 16384;          // 128*128
constexpr int TILE   = 256;            // columns staged in LDS per step (== blockDim)
constexpr int PITCH  = TILE + 1;       // +1 float: break 64-bank conflicts
constexpr int NPERM  = 40320;          // 8!
constexpr int SLICES = 8;              // HW split factor for occupancy
constexpr int SLICE_COLS = HWn / SLICES;   // 2048
constexpr int PARTF  = 160;            // per-(pair,slice) partial: 128 cross + 16 ent (+pad)

#if defined(__has_builtin)
#  if __has_builtin(__builtin_amdgcn_global_load_async_to_lds_b32)
#    define USE_ASYNC_LDS 1
#  endif
#endif
#ifndef USE_ASYNC_LDS
#  define USE_ASYNC_LDS 0
#endif

__device__ __forceinline__ void wait_asynccnt0()
{
#if defined(__has_builtin) && __has_builtin(__builtin_amdgcn_s_wait_asynccnt)
    __builtin_amdgcn_s_wait_asynccnt(0);
#else
    asm volatile("s_wait_asynccnt 0x0" ::: "memory");
#endif
}

// ---------------------------------------------------------------------------
// Kernel 1: block (pair, slice) computes over its 2048-column slice:
//   entsum[r]    = sum_hw t_r * log(t_r+eps)              (r = 2 batches x K rows)
//   cross[M][N]  = sum_hw log(p_M+eps) * t_N              (via f32 WMMA)
// WMMA packing: A(16x4) rows M=0..7 -> logp batch 2b, M=8..15 -> batch 2b+1;
//               B(4x16) cols N likewise for t. Diagonal 8x8 blocks of D valid.
// Raw partial sums go to part[pair][slice][160].
// ---------------------------------------------------------------------------
__global__ __launch_bounds__(256)
void cost_wmma_kernel(const float* __restrict__ pred,
                      const float* __restrict__ aug,
                      float* __restrict__ part)
{
    __shared__ float lA[16 * PITCH];   // log(pred+eps), 16 rows (2 batches x K)
    __shared__ float lB[16 * PITCH];   // aug rows

    const int tid  = threadIdx.x;
    const int lane = tid & 31;
    const int wv   = tid >> 5;         // wave id 0..7
    const int b0   = blockIdx.x * 2;
    const int hwbase = blockIdx.y * SLICE_COLS;

    // ISA f32 16x16x4 operand layout: lanes 0-15 hold K={0,1}, lanes 16-31 K={2,3};
    // VGPR index selects within the pair. M (for A) / N (for B) = lane & 15.
    const int m    = lane & 15;
    const int hi   = lane >> 4;        // 0 or 1
    const int koff = hi * 2;

    float entp[16];
#pragma unroll
    for (int r = 0; r < 16; ++r) entp[r] = 0.0f;

    v8f acc = {};
    const int c = tid;                 // thread t owns column t of every row

    for (int hw0 = hwbase; hw0 < hwbase + SLICE_COLS; hw0 += TILE) {
#if USE_ASYNC_LDS
        // Fire async global->LDS copies for the 16 aug rows (ASYNCcnt path,
        // data bypasses VGPRs entirely).
#pragma unroll
        for (int r = 0; r < 16; ++r) {
            const int bb = b0 + (r >> 3);
            const int kk = r & 7;
            const size_t base = ((size_t)(bb * Kn + kk)) * HWn + hw0 + c;
            __builtin_amdgcn_global_load_async_to_lds_b32(
                (gas_int*)(aug + base),
                (las_int*)(&lB[r * PITCH + c]),
                0, 0);
        }
#endif
        // pred rows must pass through VALU for the log transform.
#pragma unroll
        for (int r = 0; r < 16; ++r) {
            const int bb = b0 + (r >> 3);
            const int kk = r & 7;
            const size_t base = ((size_t)(bb * Kn + kk)) * HWn + hw0 + c;
            const float pv = pred[base];
            lA[r * PITCH + c] = __logf(pv + EPSF);
#if !USE_ASYNC_LDS
            const float tv = aug[base];
            lB[r * PITCH + c] = tv;
            entp[r] += tv * __logf(tv + EPSF);
#endif
        }
#if USE_ASYNC_LDS
        wait_asynccnt0();
        __syncthreads();
        // entropy term read back from LDS (written by the async engine)
#pragma unroll
        for (int r = 0; r < 16; ++r) {
            const float tv = lB[r * PITCH + c];
            entp[r] += tv * __logf(tv + EPSF);
        }
#else
        __syncthreads();
#endif

        // Each wave consumes its own 32-column slice: preload all operand
        // pairs, then 8 back-to-back WMMAs (one dscnt wait per tile).
        const int kstart = wv * 32;
        v2f a[8], bvv[8];
#pragma unroll
        for (int u = 0; u < 8; ++u) {
            const int k0 = kstart + u * 4 + koff;
            a[u].x   = lA[m * PITCH + k0];
            a[u].y   = lA[m * PITCH + k0 + 1];
            bvv[u].x = lB[m * PITCH + k0];
            bvv[u].y = lB[m * PITCH + k0 + 1];
        }
#pragma unroll
        for (int u = 0; u < 8; ++u) {
            // (neg_a, A, neg_b, B, c_mod, C, reuse_a, reuse_b)
            acc = __builtin_amdgcn_wmma_f32_16x16x4_f32(
                false, a[u], false, bvv[u], (short)0, acc, false, false);
        }
        __syncthreads();
    }

    // Deterministic cross-wave combine (reuse LDS tiles as scratch).
    float* accbuf = lA;   // 8 waves * 32 lanes * 8 = 2048 floats
    float* entbuf = lB;   // 16 rows * 256 threads = 4096 floats
#pragma unroll
    for (int v = 0; v < 8; ++v) accbuf[(wv * 32 + lane) * 8 + v] = acc[v];
#pragma unroll
    for (int r = 0; r < 16; ++r) entbuf[r * 256 + tid] = entp[r];
    __syncthreads();

    float* pp = part + ((size_t)blockIdx.x * SLICES + blockIdx.y) * PARTF;

    if (tid < 16) {                    // serial (deterministic) entropy reduce
        float s = 0.0f;
        for (int c2 = 0; c2 < 256; ++c2) s += entbuf[tid * 256 + c2];
        pp[128 + tid] = s;             // raw sum
    }
    if (wv == 0) {
        v8f atot = {};
#pragma unroll
        for (int v = 0; v < 8; ++v) {
            float s = 0.0f;
#pragma unroll
            for (int w2 = 0; w2 < 8; ++w2) s += accbuf[(w2 * 32 + lane) * 8 + v];
            atot[v] = s;
        }
        // D layout (16x16 f32 C/D): lane half selects M block (v or v+8), N = lane&15.
        const int n  = lane & 15;
        const int bN = n >> 3;
        if (hi == bN) {                // same-batch (diagonal) block only
            const int j = n & 7;
#pragma unroll
            for (int v = 0; v < 8; ++v) {
                pp[bN * 64 + v * 8 + j] = atot[v];   // raw sum
            }
        }
    }
}

// ---------------------------------------------------------------------------
// Kernel 2: fold the 8 HW-slices into C[b,i,j] (serial over slices -> exact,
// deterministic). tid encodes (bN, i, j).
// ---------------------------------------------------------------------------
__global__ __launch_bounds__(128)
void combine_kernel(const float* __restrict__ part, float* __restrict__ Cmat)
{
    const int pair = blockIdx.x;
    const int tid  = threadIdx.x;      // == bN*64 + i*8 + j
    const int bN = tid >> 6;
    const int j  = tid & 7;
    const float* pp = part + (size_t)pair * SLICES * PARTF;
    float cr = 0.0f, en = 0.0f;
    for (int s = 0; s < SLICES; ++s) {
        cr += pp[s * PARTF + tid];
        en += pp[s * PARTF + 128 + bN * 8 + j];
    }
    Cmat[(2 * pair + bN) * 64 + (tid & 63)] = (en - cr) * (1.0f / (float)HWn);
}

// ---------------------------------------------------------------------------
// Kernel 3: per batch, exhaustive min over 8! permutations.
// Lehmer-decode on the fly (nibble-packed remaining-element list).
// Tie-break to lowest perm index to match jnp.argmin exactly.
// Writes inv[] with pred_perm[b,i] = pred[b, inv[i]]  (inv[perm[i]] = i).
// ---------------------------------------------------------------------------
__global__ __launch_bounds__(256)
void assign_kernel(const float* __restrict__ Cmat, int* __restrict__ invperm)
{
    __shared__ float Cs[64];
    __shared__ float redV[256];
    __shared__ int   redI[256];
    const int b   = blockIdx.x;
    const int tid = threadIdx.x;
    if (tid < 64) Cs[tid] = Cmat[b * 64 + tid];
    __syncthreads();

    float bestC = 3.4e38f;
    int   bestP = 0;
    for (int p = tid; p < NPERM; p += 256) {
        unsigned av = 0x76543210u;     // remaining columns, 4 bits each
        int rem = p;
        float cost = 0.0f;
        const int fct[7] = {5040, 720, 120, 24, 6, 2, 1};
#pragma unroll
        for (int i = 0; i < 7; ++i) {
            const int d = rem / fct[i];
            rem -= d * fct[i];
            cost += Cs[i * 8 + ((av >> (4 * d)) & 0xF)];
            const unsigned lowmask = (1u << (4 * d)) - 1u;
            av = (av & lowmask) | ((av >> 4) & ~lowmask);
        }
        cost += Cs[56 + (av & 0xF)];
        if (cost < bestC) { bestC = cost; bestP = p; }  // keeps lowest p on ties
    }
    redV[tid] = bestC; redI[tid] = bestP;
    __syncthreads();
    for (int s = 128; s > 0; s >>= 1) {
        if (tid < s) {
            const float v2 = redV[tid + s]; const int i2 = redI[tid + s];
            if (v2 < redV[tid] || (v2 == redV[tid] && i2 < redI[tid])) {
                redV[tid] = v2; redI[tid] = i2;
            }
        }
        __syncthreads();
    }
    if (tid == 0) {
        int p = redI[0];
        unsigned av = 0x76543210u;
        int perm[8];
        const int fct[7] = {5040, 720, 120, 24, 6, 2, 1};
#pragma unroll
        for (int i = 0; i < 7; ++i) {
            const int d = p / fct[i];
            p -= d * fct[i];
            perm[i] = (av >> (4 * d)) & 0xF;
            const unsigned lowmask = (1u << (4 * d)) - 1u;
            av = (av & lowmask) | ((av >> 4) & ~lowmask);
        }
        perm[7] = av & 0xF;
        int inv[8];
#pragma unroll
        for (int i = 0; i < 8; ++i) inv[perm[i]] = i;
#pragma unroll
        for (int i = 0; i < 8; ++i) invperm[b * 8 + i] = inv[i];
    }
}

// ---------------------------------------------------------------------------
// Kernel 4: stream pred_perm rows to d_out[1..] and accumulate per-(b,k) KL.
// ---------------------------------------------------------------------------
__global__ __launch_bounds__(256)
void permute_loss_kernel(const float* __restrict__ pred,
                         const float* __restrict__ aug,
                         const int* __restrict__ invperm,
                         float* __restrict__ out,       // [0]=loss, [1..]=pred_perm
                         float* __restrict__ lossbuf)   // B*K partials
{
    const int bk  = blockIdx.x;
    const int b   = bk >> 3;
    const int i   = bk & 7;
    const int src = invperm[b * 8 + i];
    const float* prow = pred + ((size_t)(b * 8 + src)) * HWn;
    const float* trow = aug  + ((size_t)bk) * HWn;
    float* orow = out + 1 + (size_t)bk * HWn;

    float accl = 0.0f;
    for (int c = threadIdx.x; c < HWn; c += 256) {
        if (c + 4096 < HWn) {          // stay in-row: gfx1250 global_prefetch_b8
            __builtin_prefetch(prow + c + 4096, 0, 0);
            __builtin_prefetch(trow + c + 4096, 0, 0);
        }
        const float pv = prow[c];
        const float tv = trow[c];
        orow[c] = pv;
        accl += tv * (__logf(tv + EPSF) - __logf(pv + EPSF));
    }
    __shared__ float red[256];
    red[threadIdx.x] = accl;
    __syncthreads();
    for (int s = 128; s > 0; s >>= 1) {
        if (threadIdx.x < s) red[threadIdx.x] += red[threadIdx.x + s];
        __syncthreads();
    }
    if (threadIdx.x == 0) lossbuf[bk] = red[0] * (1.0f / (float)HWn);
}

// ---------------------------------------------------------------------------
// Kernel 5: reduce B*K=512 partials -> scalar mean at d_out[0].
// ---------------------------------------------------------------------------
__global__ __launch_bounds__(256)
void final_loss_kernel(const float* __restrict__ lossbuf, float* __restrict__ out)
{
    __shared__ float red[256];
    const int tid = threadIdx.x;
    red[tid] = lossbuf[tid] + lossbuf[tid + 256];
    __syncthreads();
    for (int s = 128; s > 0; s >>= 1) {
        if (tid < s) red[tid] += red[tid + s];
        __syncthreads();
    }
    if (tid == 0) out[0] = red[0] * (1.0f / (float)(Bn * Kn));
}

// ---------------------------------------------------------------------------
extern "C" void kernel_launch(void* const* d_in, const int* in_sizes, int n_in,
                              void* d_out, int out_size, void* d_ws, size_t ws_size,
                              hipStream_t stream)
{
    const float* pred = (const float*)d_in[0];
    const float* aug  = (const float*)d_in[1];
    float* out = (float*)d_out;

    char* ws = (char*)d_ws;
    size_t off = 0;
    float* part = (float*)(ws + off);                 // 32*8*160 f32 = 160 KB
    off += (size_t)(Bn / 2) * SLICES * PARTF * sizeof(float);
    float* Cmat = (float*)(ws + off);                 // 64*64 f32
    off += (size_t)Bn * 64 * sizeof(float);
    int* invperm = (int*)(ws + off);                  // 64*8 i32
    off += (size_t)Bn * Kn * sizeof(int);
    float* lossbuf = (float*)(ws + off);              // 512 f32

    dim3 grid1(Bn / 2, SLICES);
    cost_wmma_kernel   <<<grid1,   256, 0, stream>>>(pred, aug, part);
    combine_kernel     <<<Bn / 2,  128, 0, stream>>>(part, Cmat);
    assign_kernel      <<<Bn,      256, 0, stream>>>(Cmat, invperm);
    permute_loss_kernel<<<Bn * Kn, 256, 0, stream>>>(pred, aug, invperm, out, lossbuf);
    final_loss_kernel  <<<1,       256, 0, stream>>>(lossbuf, out);
}